// ReservoirDetector_33011118637131
// MI455X (gfx1250) — compile-verified
//
#include <hip/hip_runtime.h>
#include <math.h>

#define EPSV   1e-6f
#define BLK    256
#define NLAG   9      // order + 1
#define PORD   8      // AR order (fixed by reference: order == 8)
#define DK_ITERS 40

// ---------- CDNA5 async global->LDS staging (guarded, with fallback) ----------
#if defined(__has_builtin)
#  if __has_builtin(__builtin_amdgcn_global_load_async_to_lds_b32)
#    define HAVE_ASYNC_LDS 1
#  endif
#  if __has_builtin(__builtin_amdgcn_s_wait_asynccnt)
#    define WAIT_ASYNC() __builtin_amdgcn_s_wait_asynccnt(0)
#  else
#    define WAIT_ASYNC() asm volatile("s_wait_asynccnt 0x0" ::: "memory")
#  endif
#endif
#ifndef WAIT_ASYNC
#  define WAIT_ASYNC() asm volatile("s_wait_asynccnt 0x0" ::: "memory")
#endif

typedef __attribute__((address_space(1))) float g1f_t;
typedef __attribute__((address_space(3))) float l3f_t;
typedef __attribute__((address_space(1))) int   g1i_t;
typedef __attribute__((address_space(3))) int   l3i_t;

// One 4-byte async global->LDS copy (per-lane), tracked by ASYNCcnt.
__device__ __forceinline__ void async_copy_dword(const float* gsrc, float* ldst) {
#ifdef HAVE_ASYNC_LDS
    // two-step casts: addrspacecast, then same-AS type pun (builtin wants int*)
    g1f_t* gf = (g1f_t*)const_cast<float*>(gsrc);
    l3f_t* lf = (l3f_t*)ldst;
    __builtin_amdgcn_global_load_async_to_lds_b32((g1i_t*)gf, (l3i_t*)lf, 0, 0);
#else
    *ldst = *gsrc;
#endif
}

// ---------- tiny complex helpers (fp32, fma-based) ----------
struct cpx { float re, im; };
__device__ __forceinline__ cpx cmk(float r, float i) { cpx z; z.re = r; z.im = i; return z; }
__device__ __forceinline__ cpx csub(cpx a, cpx b) { return cmk(a.re - b.re, a.im - b.im); }
__device__ __forceinline__ cpx cmul(cpx a, cpx b) {
    return cmk(__builtin_fmaf(a.re, b.re, -(a.im * b.im)),
               __builtin_fmaf(a.re, b.im,  (a.im * b.re)));
}
__device__ __forceinline__ cpx conjc(cpx a) { return cmk(a.re, -a.im); }
__device__ __forceinline__ cpx cscale(cpx a, float s) { return cmk(a.re * s, a.im * s); }
__device__ __forceinline__ float cabs2(cpx a) { return __builtin_fmaf(a.re, a.re, a.im * a.im); }
__device__ __forceinline__ cpx cdiv(cpx a, cpx b) {
    float d   = __builtin_fmaf(b.re, b.re, b.im * b.im) + 1e-30f;
    float inv = 1.0f / d;
    return cmk((a.re * b.re + a.im * b.im) * inv,
               (a.im * b.re - a.re * b.im) * inv);
}
__device__ __forceinline__ int finite_c(cpx a) {
    return __builtin_isfinite(a.re) && __builtin_isfinite(a.im);
}

// ---------- kernel: Levinson-Durbin (order 8) + AR poles via Durand-Kerner ----------
__global__ __launch_bounds__(BLK)
void levinson_poles_kernel(const float* __restrict__ gre,
                           const float* __restrict__ gim,
                           float2* __restrict__ out_a,
                           float2* __restrict__ out_k,
                           float*  __restrict__ out_e,
                           float2* __restrict__ out_p,
                           int B)
{
    __shared__ float sre[BLK * NLAG];
    __shared__ float sim[BLK * NLAG];

    const int tid = threadIdx.x;
    const long long base  = (long long)blockIdx.x * BLK;   // first batch row of this block
    const long long gbase = base * NLAG;                   // first dword of this block's tile
    int nrows = B - (int)base; if (nrows > BLK) nrows = BLK;
    const int ndw = nrows * NLAG;

    // ---- stage this block's [rows x 9] tiles of r_re / r_im into LDS ----
#pragma unroll
    for (int i = 0; i < NLAG; ++i) {
        int idx  = tid + i * BLK;              // 0 .. BLK*9-1, always in LDS range
        int cidx = (idx < ndw) ? idx : 0;      // clamp OOB rows to a valid address
        async_copy_dword(gre + gbase + cidx, &sre[idx]);
        async_copy_dword(gim + gbase + cidx, &sim[idx]);
    }
#ifdef HAVE_ASYNC_LDS
    WAIT_ASYNC();
#endif
    __syncthreads();

    const long long b = base + tid;
    if (b >= B) return;

    // LDS row -> registers (stride 9 is odd => 32 lanes hit 32 distinct banks)
    float rre[NLAG], rim[NLAG];
#pragma unroll
    for (int j = 0; j < NLAG; ++j) {
        rre[j] = sre[tid * NLAG + j];
        rim[j] = sim[tid * NLAG + j];
    }

    // ---- Levinson-Durbin, fully unrolled (order 8) ----
    float e = fmaxf(rre[0], EPSV);
    cpx a[PORD], kap[PORD];
#pragma unroll
    for (int j = 0; j < PORD; ++j) { a[j] = cmk(0.f, 0.f); kap[j] = cmk(0.f, 0.f); }

#pragma unroll
    for (int i = 1; i <= PORD; ++i) {
        cpx acc = cmk(rre[i], rim[i]);
#pragma unroll
        for (int j = 0; j <= i - 2; ++j) {
            cpx rr = cmk(rre[i - 1 - j], rim[i - 1 - j]);
            acc = csub(acc, cmul(a[j], rr));
        }
        cpx k = cscale(acc, 1.0f / e);
        kap[i - 1] = k;

        cpx na[PORD];
#pragma unroll
        for (int j = 0; j <= i - 2; ++j) na[j] = csub(a[j], cmul(k, conjc(a[i - 2 - j])));
#pragma unroll
        for (int j = 0; j <= i - 2; ++j) a[j] = na[j];
        a[i - 1] = k;

        e = e * fmaxf(1.0f - cabs2(k), EPSV);
    }

    // ---- store a, kappa, err_var (complex interleaved re/im) ----
#pragma unroll
    for (int j = 0; j < PORD; ++j) {
        out_a[b * PORD + j] = make_float2(a[j].re, a[j].im);
        out_k[b * PORD + j] = make_float2(kap[j].re, kap[j].im);
    }
    out_e[b] = e;

    // ---- poles: roots of z^8 - a0 z^7 - ... - a7 (companion eigenvalues) ----
    cpx c[PORD];
    float mx = 0.0f;
#pragma unroll
    for (int j = 0; j < PORD; ++j) {
        cpx aj = a[j];
        if (!finite_c(aj)) aj = cmk(0.f, 0.f);   // mirrors _finite_c filtering of C
        c[j] = aj;
        mx = fmaxf(mx, fmaxf(fabsf(aj.re), fabsf(aj.im)));
    }
    const float R = 1.0f + 1.5f * mx;            // Cauchy root bound (loose)

    cpx z[PORD];
#pragma unroll
    for (int j = 0; j < PORD; ++j) {
        float th = 0.78539816339f * (float)j + 0.4f;   // 2*pi/8 spacing + offset
        z[j] = cmk(R * __cosf(th), R * __sinf(th));
    }

    const float tol2 = 1e-12f * R * R;
    for (int it = 0; it < DK_ITERS; ++it) {
        float maxd2 = 0.0f;
#pragma unroll
        for (int i = 0; i < PORD; ++i) {
            cpx zi = z[i];
            // Horner: p(z) = ((..(1*z - c0)z - c1)..)z - c7
            cpx num = cmk(1.f, 0.f);
#pragma unroll
            for (int j = 0; j < PORD; ++j) num = csub(cmul(num, zi), c[j]);
            cpx den = cmk(1.f, 0.f);
#pragma unroll
            for (int j = 0; j < PORD; ++j) {
                if (j != i) den = cmul(den, csub(zi, z[j]));
            }
            cpx d = cdiv(num, den);
            z[i] = csub(zi, d);                  // Gauss-Seidel update
            maxd2 = fmaxf(maxd2, cabs2(d));
        }
        if (!__any(maxd2 > tol2)) break;         // wave-uniform early exit
    }

#pragma unroll
    for (int j = 0; j < PORD; ++j) {
        cpx zj = z[j];
        if (!finite_c(zj)) zj = cmk(0.f, 0.f);
        out_p[b * PORD + j] = make_float2(zj.re, zj.im);
    }
}

// ---------- launch ----------
extern "C" void kernel_launch(void* const* d_in, const int* in_sizes, int n_in,
                              void* d_out, int out_size, void* d_ws, size_t ws_size,
                              hipStream_t stream) {
    const float* r_re = (const float*)d_in[0];
    const float* r_im = (const float*)d_in[1];
    // d_in[2] is `order` (==8, fixed; recursion is compile-time unrolled for 8)
    const int B = in_sizes[0] / NLAG;

    float*  out   = (float*)d_out;
    float2* out_a = (float2*)out;                          // a:      B*8 complex
    float2* out_k = (float2*)(out + (size_t)B * 16);       // kappa:  B*8 complex
    float*  out_e = out + (size_t)B * 32;                  // err_var: B real
    float2* out_p = (float2*)(out + (size_t)B * 33);       // poles:  B*8 complex

    const int grid = (B + BLK - 1) / BLK;
    levinson_poles_kernel<<<grid, BLK, 0, stream>>>(r_re, r_im, out_a, out_k, out_e, out_p, B);
}